// F1_Loss_77902116815085
// MI455X (gfx1250) — compile-verified
//
#include <hip/hip_runtime.h>

#define F1_EPS 1e-7f

typedef __attribute__((ext_vector_type(2))) float v2f;
typedef __attribute__((ext_vector_type(4))) float v4f;
typedef __attribute__((ext_vector_type(8))) float v8f;
typedef __attribute__((ext_vector_type(4))) int   v4i;

// Sum one f32 value across all 32 lanes of a wave using
// V_WMMA_F32_16X16X4_F32 with an all-ones B matrix.
//  A(16x4): lane L puts x in its K=0 (lanes 0-15) / K=2 (lanes 16-31) slot,
//           zero in the other slot  -> rowsum_m = x_m + x_{m+16}
//  B = ones (layout independent)    -> D[m,n] = rowsum_m for all n
//  D(16x16): VGPR r holds m=r (lanes 0-15) / m=r+8 (lanes 16-31), so the sum
//  of the 8 D regs covers rows 0-7 / 8-15 per lane half; one xor-16 fold
//  yields the full 32-lane sum in every lane.
__device__ __forceinline__ float wave_sum32_wmma(float x) {
    v2f a; a[0] = x;    a[1] = 0.0f;
    v2f b; b[0] = 1.0f; b[1] = 1.0f;
    v8f c = {};
    v8f d = __builtin_amdgcn_wmma_f32_16x16x4_f32(
        /*neg_a=*/false, a, /*neg_b=*/false, b,
        /*c_mod=*/(short)0, c, /*reuse_a=*/false, /*reuse_b=*/false);
    float t = ((d[0] + d[1]) + (d[2] + d[3])) + ((d[4] + d[5]) + (d[6] + d[7]));
    t += __shfl_xor(t, 16);
    return t;
}

__global__ void f1_zero_kernel(float* __restrict__ acc) {
    if (threadIdx.x < 4) acc[threadIdx.x] = 0.0f;
}

__global__ void __launch_bounds__(256)
f1_partial_kernel(const float* __restrict__ y_pred,
                  const int*   __restrict__ y_true,
                  float* __restrict__ acc, int n) {
    float s0 = 0.0f;   // sum col0
    float sa = 0.0f;   // sum col0 * t
    float sc = 0.0f;   // sum t
    const int nvec   = n >> 2;
    const int gid    = blockIdx.x * blockDim.x + threadIdx.x;
    const int stride = gridDim.x * blockDim.x;

    const v4f* __restrict__ yp4 = reinterpret_cast<const v4f*>(y_pred);
    const v4i* __restrict__ yt4 = reinterpret_cast<const v4i*>(y_true);

    // Streaming (non-temporal) b128 loads: 268 MB read once, no L2 reuse.
    for (int i = gid; i < nvec; i += stride) {
        v4f p = __builtin_nontemporal_load(yp4 + i);
        v4i t = __builtin_nontemporal_load(yt4 + i);
        float c0 = fminf(p[0], 1.0f - p[0]);
        float c1 = fminf(p[1], 1.0f - p[1]);
        float c2 = fminf(p[2], 1.0f - p[2]);
        float c3 = fminf(p[3], 1.0f - p[3]);
        float t0 = (float)t[0], t1 = (float)t[1];
        float t2 = (float)t[2], t3 = (float)t[3];
        s0 += (c0 + c1) + (c2 + c3);
        sa  = fmaf(c0, t0, fmaf(c1, t1, fmaf(c2, t2, fmaf(c3, t3, sa))));
        sc += (t0 + t1) + (t2 + t3);
    }
    // Tail (n not multiple of 4) — n is 2^25 in practice, this is just safety.
    for (int i = (nvec << 2) + gid; i < n; i += stride) {
        float p = y_pred[i];
        float c = fminf(p, 1.0f - p);
        float t = (float)y_true[i];
        s0 += c;
        sa  = fmaf(c, t, sa);
        sc += t;
    }

    // Wave-level reduction via WMMA (EXEC all-ones here: full 256-thread block,
    // reconverged after the loops).
    float w0 = wave_sum32_wmma(s0);
    float w1 = wave_sum32_wmma(sa);
    float w2 = wave_sum32_wmma(sc);

    // Block-level reduction: one slot per wave (8 waves of 32 on CDNA5).
    __shared__ float red[3][8];
    const int lane = threadIdx.x & 31;
    const int wid  = threadIdx.x >> 5;
    if (lane == 0) { red[0][wid] = w0; red[1][wid] = w1; red[2][wid] = w2; }
    __syncthreads();
    if (threadIdx.x < 3) {
        float s = 0.0f;
        const int nw = blockDim.x >> 5;
        for (int w = 0; w < nw; ++w) s += red[threadIdx.x][w];
        atomicAdd(&acc[threadIdx.x], s);   // global_atomic_add_f32, 3 per block
    }
}

__global__ void f1_finalize_kernel(const float* __restrict__ acc,
                                   float* __restrict__ out, int n) {
    if (threadIdx.x != 0 || blockIdx.x != 0) return;
    const float S0 = acc[0];   // sum col0
    const float A  = acc[1];   // sum col0*t
    const float C1 = acc[2];   // count of t==1
    const float Nf = (float)n;

    const float tp0 = S0 - A;
    const float fp0 = A;
    const float fn0 = (Nf - C1) - (S0 - A);
    const float tp1 = C1 - A;
    const float fp1 = fn0;
    const float fn1 = A;

    const float p0  = tp0 / (tp0 + fp0 + F1_EPS);
    const float r0  = tp0 / (tp0 + fn0 + F1_EPS);
    float f10 = 2.0f * (p0 * r0) / (p0 + r0 + F1_EPS);
    const float p1  = tp1 / (tp1 + fp1 + F1_EPS);
    const float r1  = tp1 / (tp1 + fn1 + F1_EPS);
    float f11 = 2.0f * (p1 * r1) / (p1 + r1 + F1_EPS);

    f10 = fminf(fmaxf(f10, F1_EPS), 1.0f - F1_EPS);
    f11 = fminf(fmaxf(f11, F1_EPS), 1.0f - F1_EPS);

    out[0] = 1.0f - 0.5f * (f10 + f11);
}

extern "C" void kernel_launch(void* const* d_in, const int* in_sizes, int n_in,
                              void* d_out, int out_size, void* d_ws, size_t ws_size,
                              hipStream_t stream) {
    const float* y_pred = (const float*)d_in[0];
    const int*   y_true = (const int*)d_in[1];
    float*       acc    = (float*)d_ws;
    float*       out    = (float*)d_out;
    const int n = in_sizes[0];

    // 1) zero the 3 accumulators (harness does not re-poison between replays)
    f1_zero_kernel<<<1, 32, 0, stream>>>(acc);

    // 2) streaming reduction: ~1M threads -> 8 b128-pair iterations each
    const int threads = 256;
    int blocks = (int)(((long long)(n >> 2) + threads - 1) / threads);
    if (blocks > 4096) blocks = 4096;
    if (blocks < 1)    blocks = 1;
    f1_partial_kernel<<<blocks, threads, 0, stream>>>(y_pred, y_true, acc, n);

    // 3) O(1) scalar finalize
    f1_finalize_kernel<<<1, 32, 0, stream>>>(acc, out, n);
}